// GLMVQ_17944373362989
// MI455X (gfx1250) — compile-verified
//
#include <hip/hip_runtime.h>
#include <math.h>

typedef __attribute__((ext_vector_type(2))) float v2f;
typedef __attribute__((ext_vector_type(8))) float v8f;

#define DD 512
#define PP 256
#define CC 8
#define BB 4096

// ---- CDNA5 async global->LDS (no VGPR round trip, ASYNCcnt tracked) -------
__device__ __forceinline__ unsigned lds_off(const void* p) {
    // generic LDS address: low 32 bits are the LDS byte offset (ISA 10.2)
    return (unsigned)(unsigned long long)p;
}
__device__ __forceinline__ void async_ld_b128(unsigned lds_byte_off, const void* g) {
    asm volatile("global_load_async_to_lds_b128 %0, %1, off"
                 :: "v"(lds_byte_off), "v"(g) : "memory");
}
__device__ __forceinline__ void wait_async0() {
    asm volatile("s_wait_asynccnt 0" ::: "memory");
}

// ---------------------------------------------------------------------------
// Kernel 1: tp[p] = omega[label[p]] @ w[p]   and  p2[p] = ||tp[p]||^2
// ---------------------------------------------------------------------------
__global__ __launch_bounds__(256) void k_tp(const float* __restrict__ protos,
                                            const float* __restrict__ omega,
                                            const int*   __restrict__ labels,
                                            float* __restrict__ tp,
                                            float* __restrict__ p2) {
    __shared__ float wsh[DD];
    __shared__ float red[256];
    const int p = blockIdx.x, t = threadIdx.x;
    wsh[t]       = protos[p * DD + t];
    wsh[t + 256] = protos[p * DD + t + 256];
    __syncthreads();
    const float* om = omega + (size_t)labels[p] * DD * DD;
    float ss = 0.f;
    for (int rep = 0; rep < 2; ++rep) {
        const int i = t + rep * 256;
        const float* row = om + (size_t)i * DD;
        float acc = 0.f;
        for (int j = 0; j < DD; j += 4) {
            float4 a = *(const float4*)(row + j);
            float4 b = *(const float4*)(wsh + j);
            acc += a.x * b.x + a.y * b.y + a.z * b.z + a.w * b.w;
        }
        tp[(size_t)p * DD + i] = acc;
        ss += acc * acc;
    }
    red[t] = ss;
    __syncthreads();
    for (int s = 128; s > 0; s >>= 1) {
        if (t < s) red[t] += red[t + s];
        __syncthreads();
    }
    if (t == 0) p2[p] = red[0];
}

// ---------------------------------------------------------------------------
// Kernel 2 (main): BM=32 fused  TX = Xtile @ omega[c]^T (f32 WMMA 16x16x4),
//   x2 = rowsum(TX^2),  S = TX @ TPc^T,  dist[b, c+8t] = x2 + p2 - 2S.
// grid = (B/32, C), block = 256 (8 waves). omega K-slab is double-buffered
// and prefetched with global_load_async_to_lds_b128.
// LDS (floats): xs 32x516 | ws0 512x20 | ws1 512x20 | tps 32x516 | S 1024 | x2 32
// ---------------------------------------------------------------------------
#define SM_XS   0
#define SM_WS0  (32 * 516)                  // 16512
#define SM_WS1  (SM_WS0 + 512 * 20)         // 26752
#define SM_TPS  (SM_WS1 + 512 * 20)         // 36992
#define SM_SLD  (SM_TPS + 32 * 516)         // 53504
#define SM_X2L  (SM_SLD + 1024)             // 54528
#define SM_TOT  (SM_X2L + 32)               // 54560 floats = 218240 bytes

__global__ __launch_bounds__(256) void k_dist(const float* __restrict__ x,
                                              const float* __restrict__ omega,
                                              const float* __restrict__ tp,
                                              const float* __restrict__ p2,
                                              float* __restrict__ dist) {
    extern __shared__ float smem[];
    float* xs  = smem + SM_XS;
    float* ws0 = smem + SM_WS0;
    float* ws1 = smem + SM_WS1;
    float* tps = smem + SM_TPS;
    float* Sld = smem + SM_SLD;
    float* x2l = smem + SM_X2L;

    const int tid  = threadIdx.x;
    const int lane = tid & 31;
    const int w    = tid >> 5;
    const int c    = blockIdx.y;
    const int b0   = blockIdx.x * 32;
    const float* og = omega + (size_t)c * DD * DD;

    // zero cross-wave reduction buffers
    Sld[tid] = 0.f; Sld[tid + 256] = 0.f; Sld[tid + 512] = 0.f; Sld[tid + 768] = 0.f;
    if (tid < 32) x2l[tid] = 0.f;

    // ---- prologue: async-stage X tile (32x512), TPc (32x512), omega slab 0
#pragma unroll
    for (int k = 0; k < 16; ++k) {
        const int f = tid + 256 * k;            // 4096 float4s
        const int row = f >> 7, c4 = f & 127;
        async_ld_b128(lds_off(xs + row * 516 + c4 * 4),
                      x + (size_t)(b0 + row) * DD + c4 * 4);
    }
#pragma unroll
    for (int k = 0; k < 16; ++k) {
        const int f = tid + 256 * k;            // 4096 float4s, proto p = c + 8*row
        const int row = f >> 7, c4 = f & 127;
        async_ld_b128(lds_off(tps + row * 516 + c4 * 4),
                      tp + (size_t)(c + 8 * row) * DD + c4 * 4);
    }
#pragma unroll
    for (int k = 0; k < 8; ++k) {
        const int f = tid + 256 * k;            // 2048 float4s: omega[.][0..15]
        const int row = f >> 2, c4 = f & 3;
        async_ld_b128(lds_off(ws0 + row * 20 + c4 * 4),
                      og + (size_t)row * DD + c4 * 4);
    }

    const int i0   = w * 64;                    // this wave's 64 output columns
    const int m    = lane & 15;                 // WMMA M/N lane index
    const int koff = (lane >> 4) * 2;           // WMMA K sub-index per lane half

    v8f a0 = {}, a1 = {}, a2 = {}, a3 = {};     // rows 0..15
    v8f a4 = {}, a5 = {}, a6 = {}, a7 = {};     // rows 16..31

    for (int it = 0; it < DD / 16; ++it) {
        const int j0 = it * 16;
        wait_async0();                          // this wave's staged data landed
        __syncthreads();                        // everyone's data landed / reads done
        if (it + 1 < DD / 16) {                 // prefetch next slab into other buffer
            float* wn = (it & 1) ? ws0 : ws1;
            const float* gsrc = og + (j0 + 16);
#pragma unroll
            for (int k = 0; k < 8; ++k) {
                const int f = tid + 256 * k;
                const int row = f >> 2, c4 = f & 3;
                async_ld_b128(lds_off(wn + row * 20 + c4 * 4),
                              gsrc + (size_t)row * DD + c4 * 4);
            }
        }
        const float* wc = (it & 1) ? ws1 : ws0;
#pragma unroll
        for (int kk = 0; kk < 16; kk += 4) {
            v2f av0 = *(const v2f*)(xs + (m)      * 516 + j0 + kk + koff);
            v2f av1 = *(const v2f*)(xs + (16 + m) * 516 + j0 + kk + koff);
            v2f bv0 = *(const v2f*)(wc + (i0      + m) * 20 + kk + koff);
            v2f bv1 = *(const v2f*)(wc + (i0 + 16 + m) * 20 + kk + koff);
            v2f bv2 = *(const v2f*)(wc + (i0 + 32 + m) * 20 + kk + koff);
            v2f bv3 = *(const v2f*)(wc + (i0 + 48 + m) * 20 + kk + koff);
            a0 = __builtin_amdgcn_wmma_f32_16x16x4_f32(false, av0, false, bv0, (short)0, a0, false, false);
            a1 = __builtin_amdgcn_wmma_f32_16x16x4_f32(false, av0, false, bv1, (short)0, a1, false, false);
            a2 = __builtin_amdgcn_wmma_f32_16x16x4_f32(false, av0, false, bv2, (short)0, a2, false, false);
            a3 = __builtin_amdgcn_wmma_f32_16x16x4_f32(false, av0, false, bv3, (short)0, a3, false, false);
            a4 = __builtin_amdgcn_wmma_f32_16x16x4_f32(false, av1, false, bv0, (short)0, a4, false, false);
            a5 = __builtin_amdgcn_wmma_f32_16x16x4_f32(false, av1, false, bv1, (short)0, a5, false, false);
            a6 = __builtin_amdgcn_wmma_f32_16x16x4_f32(false, av1, false, bv2, (short)0, a6, false, false);
            a7 = __builtin_amdgcn_wmma_f32_16x16x4_f32(false, av1, false, bv3, (short)0, a7, false, false);
        }
    }
    __syncthreads();

    // x2 partials + write TX back into xs (X tile is dead now)
    const int half8 = (lane >> 4) * 8;
#pragma unroll
    for (int r = 0; r < 8; ++r) {
        const float slo = a0[r] * a0[r] + a1[r] * a1[r] + a2[r] * a2[r] + a3[r] * a3[r];
        const float shi = a4[r] * a4[r] + a5[r] * a5[r] + a6[r] * a6[r] + a7[r] * a7[r];
        atomicAdd(&x2l[half8 + r], slo);
        atomicAdd(&x2l[16 + half8 + r], shi);
        const int mlo = half8 + r, mhi = 16 + half8 + r;
        xs[mlo * 516 + i0      + m] = a0[r];
        xs[mlo * 516 + i0 + 16 + m] = a1[r];
        xs[mlo * 516 + i0 + 32 + m] = a2[r];
        xs[mlo * 516 + i0 + 48 + m] = a3[r];
        xs[mhi * 516 + i0      + m] = a4[r];
        xs[mhi * 516 + i0 + 16 + m] = a5[r];
        xs[mhi * 516 + i0 + 32 + m] = a6[r];
        xs[mhi * 516 + i0 + 48 + m] = a7[r];
    }
    __syncthreads();

    // GEMM2: S[32x32] = TX @ TPc^T, K split across waves (64 each)
    v8f s0 = {}, s1 = {}, s2 = {}, s3 = {};
    const int kb = w * 64;
#pragma unroll
    for (int kk = 0; kk < 64; kk += 4) {
        v2f av0 = *(const v2f*)(xs + (m)      * 516 + kb + kk + koff);
        v2f av1 = *(const v2f*)(xs + (16 + m) * 516 + kb + kk + koff);
        v2f bv0 = *(const v2f*)(tps + (m)      * 516 + kb + kk + koff);
        v2f bv1 = *(const v2f*)(tps + (16 + m) * 516 + kb + kk + koff);
        s0 = __builtin_amdgcn_wmma_f32_16x16x4_f32(false, av0, false, bv0, (short)0, s0, false, false);
        s1 = __builtin_amdgcn_wmma_f32_16x16x4_f32(false, av0, false, bv1, (short)0, s1, false, false);
        s2 = __builtin_amdgcn_wmma_f32_16x16x4_f32(false, av1, false, bv0, (short)0, s2, false, false);
        s3 = __builtin_amdgcn_wmma_f32_16x16x4_f32(false, av1, false, bv1, (short)0, s3, false, false);
    }
#pragma unroll
    for (int r = 0; r < 8; ++r) {
        const int mlo = half8 + r, mhi = 16 + half8 + r;
        atomicAdd(&Sld[mlo * 32      + m], s0[r]);
        atomicAdd(&Sld[mlo * 32 + 16 + m], s1[r]);
        atomicAdd(&Sld[mhi * 32      + m], s2[r]);
        atomicAdd(&Sld[mhi * 32 + 16 + m], s3[r]);
    }
    __syncthreads();

    // dist[b0+mm][c + 8t] = x2 + p2 - 2S
#pragma unroll
    for (int rep = 0; rep < 4; ++rep) {
        const int idx = tid + rep * 256;        // 1024 entries
        const int mm = idx >> 5, t = idx & 31;
        const int p = c + 8 * t;
        dist[(size_t)(b0 + mm) * PP + p] = x2l[mm] + p2[p] - 2.0f * Sld[idx];
    }
}

// ---------------------------------------------------------------------------
// Kernel 3: sum of squares of omega (Frobenius-norm regularizer)
// ---------------------------------------------------------------------------
__global__ __launch_bounds__(256) void k_norm(const float* __restrict__ omega,
                                              float* __restrict__ accum) {
    __shared__ float red[256];
    const int g = blockIdx.x * 256 + threadIdx.x;
    float4 v = *(const float4*)(omega + (size_t)g * 4);
    red[threadIdx.x] = v.x * v.x + v.y * v.y + v.z * v.z + v.w * v.w;
    __syncthreads();
    for (int s = 128; s > 0; s >>= 1) {
        if (threadIdx.x < s) red[threadIdx.x] += red[threadIdx.x + s];
        __syncthreads();
    }
    if (threadIdx.x == 0) atomicAdd(accum + 1, red[0]);
}

// ---------------------------------------------------------------------------
// Kernel 4: per-row pos/neg mins -> sigmoid(mu), wave32 shuffle reduction
// ---------------------------------------------------------------------------
__global__ __launch_bounds__(256) void k_loss(const float* __restrict__ dist,
                                              const int*   __restrict__ y,
                                              const int*   __restrict__ labels,
                                              float* __restrict__ accum) {
    const int w = threadIdx.x >> 5, lane = threadIdx.x & 31;
    const int b = blockIdx.x * 8 + w;
    const int yb = y[b];
    float pos = 3.402823e38f, neg = 3.402823e38f;
    for (int k = 0; k < 8; ++k) {
        const int p = lane + 32 * k;
        const float dv = dist[(size_t)b * PP + p];
        if (labels[p] == yb) pos = fminf(pos, dv);
        else                 neg = fminf(neg, dv);
    }
    for (int off = 16; off > 0; off >>= 1) {
        pos = fminf(pos, __shfl_xor(pos, off, 32));
        neg = fminf(neg, __shfl_xor(neg, off, 32));
    }
    if (lane == 0) {
        const float mu  = (pos - neg) / (pos + neg);
        const float sig = 1.0f / (1.0f + expf(-mu));
        atomicAdd(accum, sig);
    }
}

__global__ void k_zero(float* a) { a[0] = 0.f; a[1] = 0.f; }

__global__ void k_fin(const float* __restrict__ accum, float* __restrict__ out) {
    out[0] = accum[0] * (1.0f / (float)BB) + 0.01f * sqrtf(accum[1]);
}

// ---------------------------------------------------------------------------
extern "C" void kernel_launch(void* const* d_in, const int* in_sizes, int n_in,
                              void* d_out, int out_size, void* d_ws, size_t ws_size,
                              hipStream_t stream) {
    const float* x      = (const float*)d_in[0];   // [B, D]
    const int*   y      = (const int*)  d_in[1];   // [B]
    const float* protos = (const float*)d_in[2];   // [P, D]
    const float* omega  = (const float*)d_in[3];   // [C, D, D]
    const int*   labels = (const int*)  d_in[4];   // [P]
    float* out = (float*)d_out;

    float* ws    = (float*)d_ws;
    float* tp    = ws;                                   // P*D
    float* p2    = ws + (size_t)PP * DD;                 // P
    float* dist  = p2 + PP;                              // B*P
    float* accum = dist + (size_t)BB * PP;               // 2

    k_zero<<<1, 1, 0, stream>>>(accum);
    k_tp<<<PP, 256, 0, stream>>>(protos, omega, labels, tp, p2);
    k_dist<<<dim3(BB / 32, CC), 256, SM_TOT * sizeof(float), stream>>>(x, omega, tp, p2, dist);
    k_norm<<<(CC * DD * DD) / (256 * 4), 256, 0, stream>>>(omega, accum);
    k_loss<<<BB / 8, 256, 0, stream>>>(dist, y, labels, accum);
    k_fin<<<1, 1, 0, stream>>>(accum, out);
}